// RFID_ST_Conv_120259084350
// MI455X (gfx1250) — compile-verified
//
#include <hip/hip_runtime.h>
#include <cstddef>

// ---------------------------------------------------------------------------
// ST-GCN forward, pruned to the needed receptive field, fp32 end-to-end.
// Heavy stages use V_WMMA_F32_16X16X4_F32. Activations are channel-last and
// tcn weights are pre-transposed to [dt][co][ci] so every WMMA operand pair
// is one 8-byte load. Temporal zero-padding is materialized in the buffers,
// eliminating bounds checks from the hot loops.
// ---------------------------------------------------------------------------

#define EPSF 1e-5f
#define NB 16
#define TT 256
#define VV 25
#define TA0 192   // g1 time window start; 64 valid slots + 4 zero pad = 68
#define W0  64
#define G1SLOTS 68
#define TA1 196   // h1 (60 slots) / g2 (60 valid + 4 zero pad = 64) start
#define W1  60
#define G2SLOTS 64
#define TA2 102   // h2/g3 window start, len 26
#define W2T 26

typedef float v2f __attribute__((ext_vector_type(2)));
typedef float v8f __attribute__((ext_vector_type(8)));

__device__ __forceinline__ float gelu_f(float x) {
  return 0.5f * x * (1.0f + erff(x * 0.70710678118654752440f));
}

#if __has_builtin(__builtin_amdgcn_wmma_f32_16x16x4_f32)
#define HAVE_WMMA_F32_4 1
#endif

// D = A(16x4) * B(4x16) + C(16x16), fp32.
//   A: lane(half,lm) elem e -> A[M=lm][K=2*half+e]
//   B: lane(half,lm) elem e -> B[K=2*half+e][N=lm]
//   C/D: elem r -> C[M=r+8*half][N=lm]
__device__ __forceinline__ v8f wmma_f32(v2f a, v2f b, v8f c) {
#ifdef HAVE_WMMA_F32_4
  return __builtin_amdgcn_wmma_f32_16x16x4_f32(false, a, false, b, (short)0, c,
                                               false, false);
#else
  const int lane = threadIdx.x & 31;
  const int hf = lane >> 4;
  const int lm = lane & 15;
#pragma unroll
  for (int k = 0; k < 4; ++k) {
    const int srcHalf = (k >> 1) << 4;
    float bk = __shfl((k & 1) ? b.y : b.x, srcHalf + lm, 32);
#pragma unroll
    for (int r = 0; r < 8; ++r) {
      int row = r + 8 * hf;
      float ak = __shfl((k & 1) ? a.y : a.x, srcHalf + row, 32);
      c[r] += ak * bk;
    }
  }
  return c;
#endif
}

// ---------------------------------------------------------------------------
// K1: data_bn. h0[n][tw][v][ci] (channel-last), t = TA0 + tw.
// ---------------------------------------------------------------------------
__global__ void k_prep(const float* __restrict__ x,
                       const float* __restrict__ dg, const float* __restrict__ db,
                       const float* __restrict__ dm, const float* __restrict__ dv,
                       float* __restrict__ h0) {
  int idx = blockIdx.x * blockDim.x + threadIdx.x;
  if (idx >= NB * W0 * VV * 3) return;
  int ci = idx % 3;
  int v = (idx / 3) % VV;
  int tw = (idx / 75) % W0;
  int n = idx / (75 * W0);
  float val = x[((n * TT + (TA0 + tw)) * VV + v) * 3 + ci];
  int vc = v * 3 + ci;
  float s = dg[vc] / sqrtf(dv[vc] + EPSF);
  h0[idx] = val * s + (db[vc] - dm[vc] * s);
}

// ---------------------------------------------------------------------------
// Weight transpose: w[co][ci][dt] -> o[dt][co][ci]
// ---------------------------------------------------------------------------
__global__ void k_wt(const float* __restrict__ w, float* __restrict__ o,
                     int CO, int CI) {
  int idx = blockIdx.x * blockDim.x + threadIdx.x;
  if (idx >= CO * CI * 9) return;
  int dt = idx % 9;
  int ci = (idx / 9) % CI;
  int co = idx / (9 * CI);
  o[((size_t)dt * CO + co) * CI + ci] = w[idx];
}

// ---------------------------------------------------------------------------
// Zero the materialized temporal padding slots.
// ---------------------------------------------------------------------------
__global__ void k_padz(float* __restrict__ base, int perN, int off, int len) {
  int idx = blockIdx.x * blockDim.x + threadIdx.x;
  int n = idx / len, r = idx % len;
  if (n < NB) base[(size_t)n * perN + off + r] = 0.f;
}

// ---------------------------------------------------------------------------
// K2: block1 gcn (cin=3) + adjacency einsum + bn1 + gelu
//     -> g1[n][tw (68 slots)][w][c64]   (VALU; cin=3 too skinny for WMMA)
// ---------------------------------------------------------------------------
__global__ __launch_bounds__(256) void k_block1_gcn(
    const float* __restrict__ h0, const float* __restrict__ A,
    const float* __restrict__ imp, const float* __restrict__ gw,
    const float* __restrict__ gb, const float* __restrict__ b1g,
    const float* __restrict__ b1b, const float* __restrict__ b1m,
    const float* __restrict__ b1v, float* __restrict__ g1) {
  __shared__ float xs[VV * 3];
  __shared__ float zs[3 * 64 * VV];
  const int tid = threadIdx.x;
  const int tw = blockIdx.x % W0;
  const int n = blockIdx.x / W0;
  if (tid < 75) xs[tid] = h0[(size_t)(n * W0 + tw) * 75 + tid];
  __syncthreads();
  for (int i = tid; i < 3 * 64 * VV; i += 256) {
    int v = i % VV;
    int c = (i / VV) % 64;
    int k = i / (VV * 64);
    int kc = k * 64 + c;
    float z = gb[kc];
#pragma unroll
    for (int ci = 0; ci < 3; ++ci) z += gw[kc * 3 + ci] * xs[v * 3 + ci];
    zs[i] = z;
  }
  __syncthreads();
  for (int i = tid; i < 64 * VV; i += 256) {
    int w = i % VV;
    int c = i / VV;
    float y = 0.f;
    for (int kv = 0; kv < 75; ++kv) {
      float aw = A[kv * VV + w] * imp[kv * VV + w];
      y += aw * zs[((kv / VV) * 64 + c) * VV + (kv % VV)];
    }
    float s = b1g[c] / sqrtf(b1v[c] + EPSF);
    y = y * s + (b1b[c] - b1m[c] * s);
    g1[((size_t)(n * G1SLOTS + tw) * VV + w) * 64 + c] = gelu_f(y);
  }
}

// ---------------------------------------------------------------------------
// K3: tcn1 implicit GEMM (M=64, K=9x64, dt-outer/ci-inner) + bn2 + gelu
//     g1[n][tt][v][ci] -> h1[n][tw1 (60)][v][co]
// ---------------------------------------------------------------------------
__global__ __launch_bounds__(256) void k_tcn1(
    const float* __restrict__ g1, const float* __restrict__ wt,  // [dt][co][ci]
    const float* __restrict__ tcb, const float* __restrict__ b2g,
    const float* __restrict__ b2b, const float* __restrict__ b2m,
    const float* __restrict__ b2v, float* __restrict__ h1) {
  const int wave = (blockIdx.x << 3) | (threadIdx.x >> 5);
  const int lane = threadIdx.x & 31;
  const int hf = lane >> 4, lm = lane & 15;
  const int ptile = wave % 94;
  const int mtile = (wave / 94) % 4;
  const int n = wave / (94 * 4);
  const int pos = ptile * 16 + lm;
  const bool pv = pos < W1 * VV;  // 1500 valid positions
  const int tw1 = pos / VV, v = pos % VV;
  const int arow = mtile * 16 + lm;
  // time slot for tap dt is tt = tw1 + dt  (ti = 196+tw1+dt-4, base TA0=192)
  const float* bbase =
      g1 + (size_t)n * (G1SLOTS * VV * 64) + (size_t)tw1 * (VV * 64) + v * 64 + 2 * hf;
  v8f acc = {};
  for (int dt = 0; dt < 9; ++dt) {
    const float* ap = wt + ((size_t)dt * 64 + arow) * 64 + 2 * hf;
    const float* bp = bbase + (size_t)dt * (VV * 64);
#pragma unroll
    for (int kk = 0; kk < 64; kk += 4) {
      v2f a = *(const v2f*)(ap + kk);
      v2f b = *(const v2f*)(bp + kk);
      acc = wmma_f32(a, b, acc);
    }
  }
  if (!pv) return;
  float* hp = h1 + ((size_t)(n * W1 + tw1) * VV + v) * 64;
#pragma unroll
  for (int r = 0; r < 8; ++r) {
    int co = mtile * 16 + r + 8 * hf;
    float s = b2g[co] / sqrtf(b2v[co] + EPSF);
    float y = (acc[r] + tcb[co]) * s + (b2b[co] - b2m[co] * s);
    hp[co] = gelu_f(y);
  }
}

// ---------------------------------------------------------------------------
// K4/K6: fused gcn (Z = X @ W^T + b) + adjacency einsum + bn1 + gelu.
// Channel-last in/out. One wave per 16-channel output tile, Z in LDS.
// ---------------------------------------------------------------------------
template <int CIN, int COUT, int TIN, int TOUT>
__global__ __launch_bounds__(128) void k_gcn(
    const float* __restrict__ xin, const float* __restrict__ A,
    const float* __restrict__ imp, const float* __restrict__ gw,
    const float* __restrict__ gb, const float* __restrict__ b1g,
    const float* __restrict__ b1b, const float* __restrict__ b1m,
    const float* __restrict__ b1v, float* __restrict__ gout) {
  __shared__ float Zs[4][32][48];
  const int wv = threadIdx.x >> 5;
  const int lane = threadIdx.x & 31;
  const int hf = lane >> 4, lm = lane & 15;
  constexpr int CGRPS = COUT / 64;
  const int cgrp = blockIdx.x % CGRPS;
  const int rest = blockIdx.x / CGRPS;
  const int tw = rest % TIN;
  const int n = rest / TIN;
  const int ctile = cgrp * 4 + wv;
  float(*Z)[48] = Zs[wv];
  // Stage 1: Z[v][k*16+j] = gb[kc] + sum_ci X[v][ci]*W[kc][ci], kc=k*COUT+ctile*16+j
  for (int mt = 0; mt < 2; ++mt) {
    const int vrow = mt * 16 + lm;  // rows 25..31 produce garbage, never read
    const float* xrow =
        xin + ((size_t)(n * TIN + tw) * VV + vrow) * CIN + 2 * hf;
    for (int nt = 0; nt < 3; ++nt) {
      const int kcbase = nt * COUT + ctile * 16;
      const float* gwp = gw + (size_t)(kcbase + lm) * CIN + 2 * hf;
      v8f acc = {};
#pragma unroll
      for (int kk = 0; kk < CIN; kk += 4) {
        v2f a = *(const v2f*)(xrow + kk);
        v2f b = *(const v2f*)(gwp + kk);
        acc = wmma_f32(a, b, acc);
      }
      float bias = gb[kcbase + lm];
#pragma unroll
      for (int r = 0; r < 8; ++r) Z[mt * 16 + r + 8 * hf][nt * 16 + lm] = acc[r] + bias;
    }
  }
  __syncthreads();
  // Stage 2: Y[w][c] = sum_{kv<75} Aw^T[w][kv] * Z[kv%25][(kv/25)*16 + j]
  for (int mt = 0; mt < 2; ++mt) {
    v8f acc = {};
    const int w = mt * 16 + lm;
    const bool wvalid = w < VV;
    for (int kk = 0; kk < 76; kk += 4) {
      int ka = kk + 2 * hf;
      v2f a, b;
      bool k0v = ka < 75, k1v = (ka + 1) < 75;
      a.x = (k0v && wvalid) ? A[ka * VV + w] * imp[ka * VV + w] : 0.f;
      a.y = (k1v && wvalid) ? A[(ka + 1) * VV + w] * imp[(ka + 1) * VV + w] : 0.f;
      b.x = k0v ? Z[ka % VV][(ka / VV) * 16 + lm] : 0.f;
      b.y = k1v ? Z[(ka + 1) % VV][((ka + 1) / VV) * 16 + lm] : 0.f;
      acc = wmma_f32(a, b, acc);
    }
    const int c = ctile * 16 + lm;
    float s = b1g[c] / sqrtf(b1v[c] + EPSF);
    float sh = b1b[c] - b1m[c] * s;
#pragma unroll
    for (int r = 0; r < 8; ++r) {
      int wr = mt * 16 + r + 8 * hf;
      if (wr < VV)
        gout[((size_t)(n * TOUT + tw) * VV + wr) * COUT + c] =
            gelu_f(acc[r] * s + sh);
    }
  }
}

// ---------------------------------------------------------------------------
// K5: tcn2 (M=128, K=9x128) + 1x1 conv residual (K=64) + BNs + gelu -> h2
// ---------------------------------------------------------------------------
__global__ __launch_bounds__(256) void k_tcn2(
    const float* __restrict__ g2, const float* __restrict__ h1,
    const float* __restrict__ wt,  // [dt][co][ci]
    const float* __restrict__ tcb, const float* __restrict__ b2g,
    const float* __restrict__ b2b, const float* __restrict__ b2m,
    const float* __restrict__ b2v, const float* __restrict__ rw,
    const float* __restrict__ rb, const float* __restrict__ rng,
    const float* __restrict__ rnb, const float* __restrict__ rnm,
    const float* __restrict__ rnv, float* __restrict__ h2) {
  const int wave = (blockIdx.x << 3) | (threadIdx.x >> 5);
  const int lane = threadIdx.x & 31;
  const int hf = lane >> 4, lm = lane & 15;
  const int ptile = wave % 41;
  const int mtile = (wave / 41) % 8;
  const int n = wave / (41 * 8);
  const int pos = ptile * 16 + lm;
  const bool pv = pos < W2T * VV;  // 650 valid positions
  const int t2w = pos / VV, v = pos % VV;
  const int arow = mtile * 16 + lm;
  // tap dt reads g2 slot tt = 2*t2w + 4 + dt   (ti = 2*(102+t2w)+dt-4, base 196)
  const float* bbase = g2 + (size_t)n * (G2SLOTS * VV * 128) +
                       (size_t)(2 * t2w + 4) * (VV * 128) + v * 128 + 2 * hf;
  v8f acc = {};
  for (int dt = 0; dt < 9; ++dt) {
    const float* ap = wt + ((size_t)dt * 128 + arow) * 128 + 2 * hf;
    const float* bp = bbase + (size_t)dt * (VV * 128);
#pragma unroll
    for (int kk = 0; kk < 128; kk += 4) {
      v2f a = *(const v2f*)(ap + kk);
      v2f b = *(const v2f*)(bp + kk);
      acc = wmma_f32(a, b, acc);
    }
  }
  // residual: h1 at t1 = 2*t2 -> slot 2*t2w + 8
  v8f racc = {};
  {
    const float* ap = rw + (size_t)arow * 64 + 2 * hf;
    const float* bp = h1 + ((size_t)(n * W1 + (2 * t2w + 8)) * VV + v) * 64 + 2 * hf;
#pragma unroll
    for (int kk = 0; kk < 64; kk += 4) {
      v2f a = *(const v2f*)(ap + kk);
      v2f b = *(const v2f*)(bp + kk);
      racc = wmma_f32(a, b, racc);
    }
  }
  if (!pv) return;
  float* hp = h2 + ((size_t)(n * W2T + t2w) * VV + v) * 128;
#pragma unroll
  for (int r = 0; r < 8; ++r) {
    int co = mtile * 16 + r + 8 * hf;
    float s2 = b2g[co] / sqrtf(b2v[co] + EPSF);
    float y = (acc[r] + tcb[co]) * s2 + (b2b[co] - b2m[co] * s2);
    float sr = rng[co] / sqrtf(rnv[co] + EPSF);
    float rr = (racc[r] + rb[co]) * sr + (rnb[co] - rnm[co] * sr);
    hp[co] = gelu_f(y + rr);
  }
}

// ---------------------------------------------------------------------------
// K7: block3 tcn, only output channel 255 + residual + final gelu -> out
// out[n][j] at (t3,v) = decode(1344+j); channel-last g3/h2 -> contiguous sums
// ---------------------------------------------------------------------------
__global__ __launch_bounds__(256) void k_final(
    const float* __restrict__ g3, const float* __restrict__ h2,
    const float* __restrict__ tcw, const float* __restrict__ tcb,
    const float* __restrict__ b2g, const float* __restrict__ b2b,
    const float* __restrict__ b2m, const float* __restrict__ b2v,
    const float* __restrict__ rw, const float* __restrict__ rb,
    const float* __restrict__ rng, const float* __restrict__ rnb,
    const float* __restrict__ rnm, const float* __restrict__ rnv,
    float* __restrict__ out) {
  const int n = blockIdx.x;
  const int j = threadIdx.x;
  const int rem = 1344 + j;
  const int t3 = rem / VV, v = rem % VV;
  const int CO = 255;
  float y = tcb[CO];
  const float* wrow = tcw + (size_t)CO * 256 * 9;
  for (int dt = 0; dt < 9; ++dt) {
    int t2 = 2 * t3 + dt - 4;  // >= 102 always
    if (t2 < 128) {
      const float* gp = g3 + ((size_t)(n * W2T + (t2 - TA2)) * VV + v) * 256;
      float s = 0.f;
      for (int ci = 0; ci < 256; ++ci) s += wrow[ci * 9 + dt] * gp[ci];
      y += s;
    }
  }
  float s = b2g[CO] / sqrtf(b2v[CO] + EPSF);
  y = y * s + (b2b[CO] - b2m[CO] * s);
  float rsum = rb[CO];
  const float* hp = h2 + ((size_t)(n * W2T + (2 * t3 - TA2)) * VV + v) * 128;
  const float* rrow = rw + CO * 128;
  for (int ci = 0; ci < 128; ++ci) rsum += rrow[ci] * hp[ci];
  float sr = rng[CO] / sqrtf(rnv[CO] + EPSF);
  rsum = rsum * sr + (rnb[CO] - rnm[CO] * sr);
  out[n * 256 + j] = gelu_f(y + rsum);
}

// ---------------------------------------------------------------------------
// Host launcher
// ---------------------------------------------------------------------------
extern "C" void kernel_launch(void* const* d_in, const int* in_sizes, int n_in,
                              void* d_out, int out_size, void* d_ws,
                              size_t ws_size, hipStream_t stream) {
  // Logical order = recursive insertion order of setup_inputs().
  // If the harness flattens with sorted dict keys (jax pytree), remap.
  static const int sorted_map[57] = {
      56, 0,  2,  1,  3,  4,
      15, 14, 13, 6,  5,  7,  8,  17, 16, 10, 9,  11, 12,
      28, 27, 26, 19, 18, 20, 21, 36, 35, 23, 22, 24, 25, 34, 29, 31, 30, 32, 33,
      47, 46, 45, 38, 37, 39, 40, 55, 54, 42, 41, 43, 44, 53, 48, 50, 49, 51, 52};
  const bool insertion = (n_in > 0 && in_sizes[0] == NB * TT * VV * 3);
  const float* in[57];
  for (int i = 0; i < 57; ++i)
    in[i] = (const float*)d_in[insertion ? i : sorted_map[i]];

  float* ws = (float*)d_ws;
  float* h0 = ws;                  // 16*64*75        =    76,800
  float* g1 = h0 + 76800;          // 16*68*25*64     = 1,740,800
  float* h1 = g1 + 1740800;        // 16*60*25*64     = 1,536,000
  float* g2 = h1 + 1536000;        // 16*64*25*128    = 3,276,800
  float* h2 = g2 + 3276800;        // 16*26*25*128    = 1,331,200
  float* g3 = h2 + 1331200;        // 16*26*25*256    = 2,662,400
  float* wt1 = g3 + 2662400;       // 9*64*64         =    36,864
  float* wt2 = wt1 + 36864;        // 9*128*128       =   147,456
  (void)ws_size;                   // total 10,808,320 floats (~43.2 MB)
  float* out = (float*)d_out;
  (void)out_size;

  k_prep<<<300, 256, 0, stream>>>(in[0], in[2], in[3], in[4], in[5], h0);
  k_wt<<<144, 256, 0, stream>>>(in[13], wt1, 64, 64);
  k_wt<<<576, 256, 0, stream>>>(in[26], wt2, 128, 128);
  k_padz<<<400, 256, 0, stream>>>(g1, G1SLOTS * VV * 64, W0 * VV * 64,
                                  4 * VV * 64);
  k_padz<<<800, 256, 0, stream>>>(g2, G2SLOTS * VV * 128, W1 * VV * 128,
                                  4 * VV * 128);
  k_block1_gcn<<<NB * W0, 256, 0, stream>>>(h0, in[1], in[6], in[7], in[8],
                                            in[9], in[10], in[11], in[12], g1);
  k_tcn1<<<752, 256, 0, stream>>>(g1, wt1, in[14], in[15], in[16], in[17],
                                  in[18], h1);
  k_gcn<64, 128, 60, 64><<<NB * W1 * 2, 128, 0, stream>>>(
      h1, in[1], in[19], in[20], in[21], in[22], in[23], in[24], in[25], g2);
  k_tcn2<<<656, 256, 0, stream>>>(g2, h1, wt2, in[27], in[28], in[29], in[30],
                                  in[31], in[32], in[33], in[34], in[35],
                                  in[36], in[37], h2);
  k_gcn<128, 256, 26, 26><<<NB * W2T * 4, 128, 0, stream>>>(
      h2, in[1], in[38], in[39], in[40], in[41], in[42], in[43], in[44], g3);
  k_final<<<NB, 256, 0, stream>>>(g3, h2, in[45], in[46], in[47], in[48],
                                  in[49], in[50], in[51], in[52], in[53],
                                  in[54], in[55], in[56], out);
}